// MultiPrototypeModule_37812892074189
// MI455X (gfx1250) — compile-verified
//
#include <hip/hip_runtime.h>

#define D        768
#define NCLS     64
#define KP       3
#define CHUNK    256
#define TILE     16
#define SEG_PTS  128
#define MAXSEG   32
#define KSTEPS   24          // (96 K-values per wave) / 4 per WMMA

typedef __attribute__((ext_vector_type(2))) float v2f;
typedef __attribute__((ext_vector_type(8))) float v8f;

// ---------------- bucketing (order-preserving, deterministic) ----------------

__global__ void hist_kernel(const int* __restrict__ labels, int N,
                            int* __restrict__ chunk_counts) {
  __shared__ int cnt[NCLS];
  const int t = threadIdx.x, chunk = blockIdx.x;
  if (t < NCLS) cnt[t] = 0;
  __syncthreads();
  const int n = chunk * CHUNK + t;
  if (n < N) atomicAdd(&cnt[labels[n]], 1);
  __syncthreads();
  if (t < NCLS) chunk_counts[chunk * NCLS + t] = cnt[t];
}

__global__ void scan_kernel(const int* __restrict__ chunk_counts, int nchunks,
                            int* __restrict__ chunk_base, int* __restrict__ cls_off) {
  const int c = threadIdx.x;                       // 64 threads
  int run = 0;
  for (int ch = 0; ch < nchunks; ch++) {
    chunk_base[ch * NCLS + c] = run;
    run += chunk_counts[ch * NCLS + c];
  }
  __shared__ int tot[NCLS];
  tot[c] = run;
  __syncthreads();
  if (c == 0) {
    int acc = 0;
    for (int i = 0; i < NCLS; i++) { cls_off[i] = acc; acc += tot[i]; }
    cls_off[NCLS] = acc;
  }
}

__global__ void scatter_kernel(const int* __restrict__ labels, int N,
                               const int* __restrict__ chunk_base,
                               const int* __restrict__ cls_off,
                               int* __restrict__ cls_idx) {
  __shared__ int lab[CHUNK];
  const int t = threadIdx.x, chunk = blockIdx.x;
  const int n = chunk * CHUNK + t;
  lab[t] = (n < N) ? labels[n] : -1;
  __syncthreads();
  if (n < N) {
    const int c = lab[t];
    int rank = 0;
    for (int j = 0; j < t; j++) rank += (lab[j] == c);
    cls_idx[cls_off[c] + chunk_base[chunk * NCLS + c] + rank] = n;
  }
}

// ---------------- init: seed centroids = first 3 class members ---------------

__global__ void init_kernel(const float* __restrict__ feat,
                            const int* __restrict__ cls_idx,
                            const int* __restrict__ cls_off, int N,
                            float* __restrict__ cent, float* __restrict__ csq) {
  const int c = blockIdx.x / KP, s = blockIdx.x % KP, tid = threadIdx.x;
  int pos = cls_off[c] + s;
  if (pos > N - 1) pos = N - 1;                    // safety only
  const int n = cls_idx[pos];
  const float* src = feat + (size_t)n * D;
  float* dst = cent + ((size_t)(c * KP + s)) * D;
  float local = 0.0f;
  for (int d = tid; d < D; d += 256) { const float v = src[d]; dst[d] = v; local += v * v; }
  __shared__ float red[256];
  red[tid] = local; __syncthreads();
  for (int st = 128; st > 0; st >>= 1) { if (tid < st) red[tid] += red[tid + st]; __syncthreads(); }
  if (tid == 0) csq[c * KP + s] = red[0];
}

// ---------------- assignment + deterministic partial sums (WMMA fp32) --------

__global__ void __launch_bounds__(256) assign_kernel(
    const float* __restrict__ feat, const int* __restrict__ cls_idx,
    const int* __restrict__ cls_off, const float* __restrict__ cent,
    const float* __restrict__ csq, float* __restrict__ psum,
    int* __restrict__ pcount) {
  __shared__ __align__(16) float tile_s[TILE][D + 4];  // pad: kill A-frag bank conflicts
  __shared__ float pdots[8][16][17];
  __shared__ float score_s[16][KP];
  __shared__ int assign_s[TILE];
  __shared__ int cnt_s[KP];

  const int c = blockIdx.y, seg = blockIdx.x, tid = threadIdx.x;
  const int off = cls_off[c];
  const int size = cls_off[c + 1] - off;
  const int start = seg * SEG_PTS;
  if (start >= size) return;                       // block-uniform exit
  const int npts = min(SEG_PTS, size - start);

  if (tid < KP) cnt_s[tid] = 0;

  const int wave = tid >> 5, lane = tid & 31;
  const int halfsel = lane >> 4, mrow = lane & 15;
  const int kb0 = wave * (D / 8);                  // 96 K-values per wave

  float cq[KP];
  #pragma unroll
  for (int s = 0; s < KP; s++) cq[s] = csq[c * KP + s];

  // Preload per-wave B fragments: B[K][N] = cent[N][ka], cols >= 3 are zero.
  v2f barr[KSTEPS];
  {
    const int scl = (mrow < KP) ? mrow : 0;
    const float* crow = cent + ((size_t)(c * KP + scl)) * D;
    #pragma unroll
    for (int i = 0; i < KSTEPS; i++) {
      const int ka = kb0 + i * 4 + 2 * halfsel;
      v2f b = *(const v2f*)(crow + ka);
      if (mrow >= KP) { b.x = 0.0f; b.y = 0.0f; }
      barr[i] = b;
    }
  }

  float racc[KP][D / 256];
  #pragma unroll
  for (int s = 0; s < KP; s++)
    #pragma unroll
    for (int j = 0; j < D / 256; j++) racc[s][j] = 0.0f;

  __syncthreads();

  for (int t0 = 0; t0 < npts; t0 += TILE) {
    const int valid = min(TILE, npts - t0);

    // stage 16 rows, coalesced float4
    for (int q = tid; q < TILE * (D / 4); q += 256) {
      const int r = q / (D / 4), d4 = q % (D / 4);
      float4 v;
      if (r < valid) {
        const int n = cls_idx[off + start + t0 + r];
        v = ((const float4*)(feat + (size_t)n * D))[d4];
      } else {
        v = make_float4(0.f, 0.f, 0.f, 0.f);
      }
      ((float4*)&tile_s[r][0])[d4] = v;
    }
    __syncthreads();

    // 24 chained fp32 WMMAs per wave over its K-partition
    v8f cf = {};
    #pragma unroll
    for (int i = 0; i < KSTEPS; i++) {
      const int ka = kb0 + i * 4 + 2 * halfsel;
      const v2f a = *(const v2f*)(&tile_s[mrow][ka]);
      cf = __builtin_amdgcn_wmma_f32_16x16x4_f32(
          false, a, false, barr[i], (short)0, cf, false, false);
    }
    #pragma unroll
    for (int v = 0; v < 8; v++)
      pdots[wave][v + 8 * halfsel][mrow] = cf[v];   // D-matrix: M=v+8*(l/16), N=l%16
    __syncthreads();

    // fold wave partials: score = |c|^2 - 2*dot  (same ordering as d2)
    if (tid < TILE * KP) {
      const int p = tid / KP, s = tid % KP;
      float dot = 0.0f;
      #pragma unroll
      for (int w = 0; w < 8; w++) dot += pdots[w][p][s];
      score_s[p][s] = cq[s] - 2.0f * dot;
    }
    __syncthreads();
    if (tid < TILE) {
      int best = 0; float bs = score_s[tid][0];
      #pragma unroll
      for (int s = 1; s < KP; s++) {                // first-wins ties == argmin
        const float v = score_s[tid][s];
        if (v < bs) { bs = v; best = s; }
      }
      assign_s[tid] = (tid < valid) ? best : KP;
    }
    __syncthreads();

    // deterministic accumulation: thread owns d = tid, tid+256, tid+512
    #pragma unroll
    for (int j = 0; j < D / 256; j++) {
      const int d = tid + j * 256;
      float a0 = racc[0][j], a1 = racc[1][j], a2 = racc[2][j];
      #pragma unroll
      for (int p = 0; p < TILE; p++) {
        const int s = assign_s[p];
        const float fv = tile_s[p][d];
        a0 += (s == 0) ? fv : 0.0f;
        a1 += (s == 1) ? fv : 0.0f;
        a2 += (s == 2) ? fv : 0.0f;
      }
      racc[0][j] = a0; racc[1][j] = a1; racc[2][j] = a2;
    }
    if (tid < KP) {
      int cc = 0;
      #pragma unroll
      for (int p = 0; p < TILE; p++) cc += (assign_s[p] == tid);
      cnt_s[tid] += cc;
    }
    __syncthreads();
  }

  const size_t base = (((size_t)c * MAXSEG + seg) * KP) * D;
  #pragma unroll
  for (int s = 0; s < KP; s++)
    #pragma unroll
    for (int j = 0; j < D / 256; j++)
      psum[base + (size_t)s * D + tid + j * 256] = racc[s][j];
  if (tid < KP) pcount[(c * MAXSEG + seg) * KP + tid] = cnt_s[tid];
}

// ---------------- centroid update (fixed-order segment fold) -----------------

__global__ void __launch_bounds__(256) reduce_kernel(
    const float* __restrict__ psum, const int* __restrict__ pcount,
    const int* __restrict__ cls_off, const float* __restrict__ cent_in,
    float* __restrict__ cent_out, float* __restrict__ csq_out) {
  const int c = blockIdx.x / KP, s = blockIdx.x % KP, tid = threadIdx.x;
  const int size = cls_off[c + 1] - cls_off[c];
  const int nseg = min((size + SEG_PTS - 1) / SEG_PTS, MAXSEG);

  int count = 0;
  for (int g = 0; g < nseg; g++) count += pcount[(c * MAXSEG + g) * KP + s];

  float localsq = 0.0f;
  for (int j = 0; j < D / 256; j++) {
    const int d = tid + j * 256;
    float sum = 0.0f;
    for (int g = 0; g < nseg; g++)
      sum += psum[(((size_t)c * MAXSEG + g) * KP + s) * D + d];
    float outv;
    if (count > 0) outv = sum / (float)count;      // sums / max(counts,1)
    else outv = cent_in[((size_t)(c * KP + s)) * D + d];
    cent_out[((size_t)(c * KP + s)) * D + d] = outv;
    localsq += outv * outv;
  }
  __shared__ float red[256];
  red[tid] = localsq; __syncthreads();
  for (int st = 128; st > 0; st >>= 1) { if (tid < st) red[tid] += red[tid + st]; __syncthreads(); }
  if (tid == 0) csq_out[c * KP + s] = red[0];
}

__global__ void write_labels_kernel(float* __restrict__ out_labels) {
  const int i = threadIdx.x;
  if (i < NCLS * KP) out_labels[i] = (float)(i / KP);
}

// ---------------- driver -----------------------------------------------------

extern "C" void kernel_launch(void* const* d_in, const int* in_sizes, int n_in,
                              void* d_out, int out_size, void* d_ws, size_t ws_size,
                              hipStream_t stream) {
  (void)n_in; (void)out_size; (void)ws_size;
  const float* feat = (const float*)d_in[0];
  const int* labels = (const int*)d_in[1];
  const int N = in_sizes[0] / D;
  const int nchunks = (N + CHUNK - 1) / CHUNK;

  char* ws = (char*)d_ws;
  size_t o = 0;
  auto take = [&](size_t bytes) -> void* {
    void* p = ws + o;
    o += (bytes + 255) & ~(size_t)255;
    return p;
  };
  int*   cls_idx      = (int*)take((size_t)N * 4);
  int*   chunk_counts = (int*)take((size_t)nchunks * NCLS * 4);
  int*   chunk_base   = (int*)take((size_t)nchunks * NCLS * 4);
  int*   cls_off      = (int*)take((NCLS + 1) * 4);
  float* cbuf0        = (float*)take((size_t)NCLS * KP * D * 4);
  float* cbuf1        = (float*)take((size_t)NCLS * KP * D * 4);
  float* csq          = (float*)take((size_t)NCLS * KP * 4);
  float* psum         = (float*)take((size_t)NCLS * MAXSEG * KP * D * 4);
  int*   pcount       = (int*)take((size_t)NCLS * MAXSEG * KP * 4);

  hist_kernel<<<nchunks, CHUNK, 0, stream>>>(labels, N, chunk_counts);
  scan_kernel<<<1, NCLS, 0, stream>>>(chunk_counts, nchunks, chunk_base, cls_off);
  scatter_kernel<<<nchunks, CHUNK, 0, stream>>>(labels, N, chunk_base, cls_off, cls_idx);
  init_kernel<<<NCLS * KP, 256, 0, stream>>>(feat, cls_idx, cls_off, N, cbuf0, csq);

  float* cur = cbuf0;
  float* nxt = cbuf1;
  for (int it = 0; it < 10; it++) {
    assign_kernel<<<dim3(MAXSEG, NCLS), 256, 0, stream>>>(
        feat, cls_idx, cls_off, cur, csq, psum, pcount);
    float* outc = (it == 9) ? (float*)d_out : nxt;
    reduce_kernel<<<NCLS * KP, 256, 0, stream>>>(psum, pcount, cls_off, cur, outc, csq);
    float* t = cur; cur = nxt; nxt = t;
  }
  write_labels_kernel<<<1, 256, 0, stream>>>((float*)d_out + (size_t)NCLS * KP * D);
}